// VectorQuantizer_78297253806628
// MI455X (gfx1250) — compile-verified
//
#include <hip/hip_runtime.h>

typedef __attribute__((ext_vector_type(2))) float v2f;
typedef __attribute__((ext_vector_type(4))) float v4f;
typedef __attribute__((ext_vector_type(8))) float v8f;

#define EMB_K 512
#define EMB_D 64
#define LDS_T_STRIDE 68   // transposed codebook [512][68]: rows 272B (16B-aligned), 68%64==4
                          // -> b64 B-fragment loads tile all 64 banks exactly once
#define VQ_BETA 0.25f

__global__ void vq_zero_loss(float* p) {
    if (threadIdx.x == 0 && blockIdx.x == 0) *p = 0.0f;
}

__global__ __launch_bounds__(256) void vq_kernel(
    const float* __restrict__ x,      // [N, 64] row-major
    const float* __restrict__ E,      // [64, 512] row-major (D x K)
    float* __restrict__ qOut,         // [N, 64]
    float* __restrict__ lossOut,      // [1]
    float* __restrict__ idxOut,       // [N] (indices stored as float)
    int nRows, int nPairs, int totalWaves)
{
    extern __shared__ float smem[];
    float* eLds  = smem;                          // transposed codebook [512][68]
    float* colsq = smem + EMB_K * LDS_T_STRIDE;   // [512]

    const int tid  = threadIdx.x;
    const int lane = tid & 31;
    const int wave = tid >> 5;
    const int half = lane >> 4;     // 0: lanes 0-15, 1: lanes 16-31
    const int l16  = lane & 15;

    // ---- stage codebook into LDS, transposed: eLds[n*68 + d] = E[d*512 + n] ----
    // global reads coalesced (n fastest across lanes); LDS writes only 2-way conflicted.
    for (int i = tid; i < EMB_D * EMB_K; i += 256) {
        int d = i >> 9;          // i / 512
        int n = i & (EMB_K - 1); // i % 512
        eLds[n * LDS_T_STRIDE + d] = E[i];
    }
    __syncthreads();

    // ---- per-column squared norms ||e_n||^2 (contiguous v4f reads per column) ----
    for (int n = tid; n < EMB_K; n += 256) {
        const float* row = &eLds[n * LDS_T_STRIDE];
        float s = 0.0f;
        #pragma unroll
        for (int d4 = 0; d4 < EMB_D / 4; ++d4) {
            v4f e = *(const v4f*)(row + 4 * d4);
            s += e.x * e.x + e.y * e.y + e.z * e.z + e.w * e.w;
        }
        colsq[n] = s;
    }
    __syncthreads();

    const int gwave = blockIdx.x * 8 + wave;
    float lossAcc = 0.0f;

    // Each wave owns a pair of 16-row tiles (M=32): one B fragment feeds two WMMAs.
    for (int pair = gwave; pair < nPairs; pair += totalWaves) {
        const int rowBase = pair * 32;

        // prefetch next iteration's A block (partial coverage is fine)
        const int nextBase = rowBase + totalWaves * 32;
        if (nextBase < nRows)
            __builtin_prefetch(&x[(size_t)(nextBase + lane) * EMB_D], 0, 1);

        // ---- A fragments: ISA 16x4 f32 layout, 16 K-steps cover D=64 ----
        // lane (half=0): row=l16, K = 4t+{0,1}; lane (half=1): row=l16, K = 4t+{2,3}
        v2f a0[16], a1[16];
        const float* xr0 = x + (size_t)(rowBase + l16) * EMB_D + 2 * half;
        const float* xr1 = xr0 + 16 * EMB_D;
        #pragma unroll
        for (int t = 0; t < 16; ++t) {
            a0[t] = *(const v2f*)(xr0 + 4 * t);
            a1[t] = *(const v2f*)(xr1 + 4 * t);
        }

        // ---- per-row squared norms ||x_r||^2 ----
        float p0 = 0.0f, p1 = 0.0f;
        #pragma unroll
        for (int t = 0; t < 16; ++t) {
            p0 += a0[t].x * a0[t].x + a0[t].y * a0[t].y;
            p1 += a1[t].x * a1[t].x + a1[t].y * a1[t].y;
        }
        float rs0 = p0 + __shfl_xor(p0, 16, 32);   // lanes r, r+16 cover disjoint K
        float rs1 = p1 + __shfl_xor(p1, 16, 32);
        // C-matrix VGPR v maps to row (v + 8*half): pre-broadcast needed values
        float rowsq0[8], rowsq1[8];
        #pragma unroll
        for (int v = 0; v < 8; ++v) {
            rowsq0[v] = __shfl(rs0, v + 8 * half, 32);
            rowsq1[v] = __shfl(rs1, v + 8 * half, 32);
        }

        float bestD0[8], bestD1[8];
        int   bestI0[8], bestI1[8];
        #pragma unroll
        for (int v = 0; v < 8; ++v) {
            bestD0[v] = 3.402823466e38f; bestI0[v] = 0;
            bestD1[v] = 3.402823466e38f; bestI1[v] = 0;
        }

        // ---- sweep the 32 column tiles of the codebook ----
        for (int c = 0; c < EMB_K / 16; ++c) {
            const int n = c * 16 + l16;
            // transposed B fragment base: eLds[n][2*half + 4t .. +1] -> one b64 per k-step
            const float* bBase = &eLds[n * LDS_T_STRIDE + 2 * half];
            v8f acc0 = {}, acc1 = {};
            #pragma unroll
            for (int t = 0; t < 16; ++t) {
                v2f b = *(const v2f*)(bBase + 4 * t);
                acc0 = __builtin_amdgcn_wmma_f32_16x16x4_f32(
                    false, a0[t], false, b, (short)0, acc0, false, false);
                acc1 = __builtin_amdgcn_wmma_f32_16x16x4_f32(
                    false, a1[t], false, b, (short)0, acc1, false, false);
            }
            float cs = colsq[n];
            #pragma unroll
            for (int v = 0; v < 8; ++v) {
                float d0 = rowsq0[v] - 2.0f * acc0[v] + cs;
                float d1 = rowsq1[v] - 2.0f * acc1[v] + cs;
                if (d0 < bestD0[v]) { bestD0[v] = d0; bestI0[v] = n; }
                if (d1 < bestD1[v]) { bestD1[v] = d1; bestI1[v] = n; }
            }
        }

        // ---- argmin reduction across the 16 lanes of each half-wave ----
        int iAll[32];
        #pragma unroll
        for (int v = 0; v < 8; ++v) {
            float d0 = bestD0[v]; int i0 = bestI0[v];
            float d1 = bestD1[v]; int i1 = bestI1[v];
            #pragma unroll
            for (int m = 1; m <= 8; m <<= 1) {
                float dd0 = __shfl_xor(d0, m, 32); int ii0 = __shfl_xor(i0, m, 32);
                float dd1 = __shfl_xor(d1, m, 32); int ii1 = __shfl_xor(i1, m, 32);
                if (dd0 < d0 || (dd0 == d0 && ii0 < i0)) { d0 = dd0; i0 = ii0; }
                if (dd1 < d1 || (dd1 == d1 && ii1 < i1)) { d1 = dd1; i1 = ii1; }
            }
            if (lane == 0)  { idxOut[rowBase + v]          = (float)i0;
                              idxOut[rowBase + 16 + v]     = (float)i1; }
            if (lane == 16) { idxOut[rowBase + v + 8]      = (float)i0;
                              idxOut[rowBase + 16 + v + 8] = (float)i1; }
            iAll[v]          = __shfl(i0, 0, 32);
            iAll[v + 8]      = __shfl(i0, 16, 32);
            iAll[16 + v]     = __shfl(i1, 0, 32);
            iAll[16 + v + 8] = __shfl(i1, 16, 32);
        }

        // ---- quantized output (coalesced b64) + commitment-loss accumulation ----
        #pragma unroll
        for (int r = 0; r < 32; ++r) {
            int    iv   = iAll[r];                       // wave-uniform per row
            size_t base = (size_t)(rowBase + r) * EMB_D + 2 * lane;
            v2f q  = *(const v2f*)&eLds[iv * LDS_T_STRIDE + 2 * lane];  // conflict-free b64
            v2f xv = *(const v2f*)&x[base];
            *(v2f*)&qOut[base] = q;
            float e0 = q.x - xv.x, e1 = q.y - xv.y;
            lossAcc += e0 * e0 + e1 * e1;
        }
    }

    // ---- loss: wave reduce, then one atomic per wave ----
    #pragma unroll
    for (int m = 1; m < 32; m <<= 1)
        lossAcc += __shfl_xor(lossAcc, m, 32);
    if (lane == 0)
        atomicAdd(lossOut, lossAcc * (VQ_BETA / ((float)nRows * (float)EMB_D)));
}

extern "C" void kernel_launch(void* const* d_in, const int* in_sizes, int n_in,
                              void* d_out, int out_size, void* d_ws, size_t ws_size,
                              hipStream_t stream) {
    (void)n_in; (void)out_size; (void)d_ws; (void)ws_size;
    const float* x = (const float*)d_in[0];   // [32,64,64,64] f32
    const float* E = (const float*)d_in[1];   // [64,512] f32

    const int nElems = in_sizes[0];           // N * 64
    const int nRows  = nElems / EMB_D;        // 131072
    const int nPairs = nRows / 32;            // 4096 pairs of 16-row tiles

    float* qOut    = (float*)d_out;           // [N,64]
    float* lossOut = qOut + nElems;           // [1]
    float* idxOut  = lossOut + 1;             // [N]

    vq_zero_loss<<<1, 32, 0, stream>>>(lossOut);

    const int blocks     = 256;
    const int totalWaves = blocks * 8;        // 256-thread blocks = 8 wave32s
    const size_t shmem   = (size_t)(EMB_K * LDS_T_STRIDE + EMB_K) * sizeof(float);
    vq_kernel<<<blocks, 256, shmem, stream>>>(x, E, qOut, lossOut, idxOut,
                                              nRows, nPairs, totalWaves);
}